// GCN_Model_67078799229038
// MI455X (gfx1250) — compile-verified
//
#include <hip/hip_runtime.h>
#include <math.h>

// ============================================================================
// GCN_Model forward for MI455X (gfx1250, wave32, WMMA).
// Roofline: ~7 GFLOP, ~80MB intermediates -> bandwidth bound (~3.5us floor at
// 23.3 TB/s). Strategy: fp32 end-to-end, all heavy einsums / 1x1 convs as
// V_WMMA_F32_16X16X4_F32 tiles with fused BN/ReLU/residual epilogues; raw
// reshape reinterpretations exploited so transformers read buffers flat.
// Workspace requirement: 19,783,680 floats (~79.2 MB).
// ============================================================================

#define EPS 1e-5f

typedef __attribute__((ext_vector_type(2))) float v2f;
typedef __attribute__((ext_vector_type(8))) float v8f;

__constant__ int c_part_off[6] = {0, 5, 8, 11, 14, 17};
__constant__ int c_part_idx[17] = {0,7,8,9,10, 4,5,6, 1,2,3, 11,12,13, 14,15,16};
__constant__ int c_j2p[17] = {0,2,2,2,1,1,1,0,0,0,0,3,3,3,4,4,4};

__device__ __forceinline__ float bnf(float x, const float* g, const float* b,
                                     const float* m, const float* v, int c) {
  return (x - m[c]) / sqrtf(v[c] + EPS) * g[c] + b[c];
}

// ---------------------------------------------------------------------------
// K1: bn0 + repack. X0[n][c][t*17+v] = BN(x[n,c,t,v], ch=v*2+c)
// ---------------------------------------------------------------------------
__global__ void bn0_kernel(const float* __restrict__ x, const float* g, const float* b,
                           const float* m, const float* v, float* __restrict__ X0) {
  int i = blockIdx.x * blockDim.x + threadIdx.x;
  if (i >= 1024 * 170) return;
  int n = i / 170, r = i % 170;
  int c = r / 85, tv = r % 85;
  int t = tv / 17, vv = tv % 17;
  int ch = vv * 2 + c;
  float val = x[((n * 2 + c) * 5 + t) * 17 + vv];
  X0[i] = (val - m[ch]) / sqrtf(v[ch] + EPS) * g[ch] + b[ch];
}

// ---------------------------------------------------------------------------
// K2: g0 gcn conv1x1 (2 -> 64), write GEMM-ready Y0[(n*16+c)][k*85+tv]
// ---------------------------------------------------------------------------
__global__ void gcn0_kernel(const float* __restrict__ X0, const float* __restrict__ w,
                            const float* __restrict__ bias, float* __restrict__ Y0) {
  int i = blockIdx.x * blockDim.x + threadIdx.x;
  if (i >= 1024 * 85) return;
  int n = i / 85, tv = i % 85;
  float x0 = X0[n * 170 + tv];
  float x1 = X0[n * 170 + 85 + tv];
  for (int o = 0; o < 64; ++o) {
    float val = bias[o] + x0 * w[o * 2] + x1 * w[o * 2 + 1];
    int k = o >> 4, c = o & 15;
    Y0[((long)(n * 16 + c)) * 340 + k * 85 + tv] = val;
  }
}

// ---------------------------------------------------------------------------
// WMMA GEMM for the graph einsum:  out[n][ch][w] = relu(bn1( Y[n-rows] @ Aflat ))
// Y: (N*RPN, Kd) row-major; Aflat: (Kd, W). One wave -> one 16x16 tile.
// grid = (N, MT*NT), block = 32.
// ---------------------------------------------------------------------------
__global__ __launch_bounds__(32) void einsum_wmma(
    const float* __restrict__ Y, const float* __restrict__ Bm,
    const float* __restrict__ bg, const float* __restrict__ bb,
    const float* __restrict__ bm, const float* __restrict__ bv,
    float* __restrict__ out, int RPN, int Kd, int W, int NT) {
  int n = blockIdx.x;
  int mt = blockIdx.y / NT, nt = blockIdx.y % NT;
  int lane = threadIdx.x & 31;
  int l16 = lane & 15;
  int hi = lane >> 4;
  int kh = hi * 2;
  long row = (long)n * RPN + mt * 16 + l16;
  int col = nt * 16 + l16;
  bool cok = col < W;
  int colc = cok ? col : 0;
  const float* yrow = Y + row * Kd;
  v8f acc = {};
  for (int kk = 0; kk < Kd; kk += 4) {
    v2f a, b;
    a.x = yrow[kk + kh];
    a.y = yrow[kk + kh + 1];
    float b0 = Bm[(kk + kh) * W + colc];
    float b1 = Bm[(kk + kh + 1) * W + colc];
    b.x = cok ? b0 : 0.f;
    b.y = cok ? b1 : 0.f;
    acc = __builtin_amdgcn_wmma_f32_16x16x4_f32(false, a, false, b, (short)0, acc,
                                                false, false);
  }
  if (cok) {
    for (int r = 0; r < 8; ++r) {
      int ch = mt * 16 + r + 8 * hi;
      float z = bnf(acc[r], bg, bb, bm, bv, ch);
      out[((long)n * RPN + ch) * W + col] = fmaxf(z, 0.f);
    }
  }
}

// ---------------------------------------------------------------------------
// WMMA GEMM for tcn 1x1 conv: out[n][o][w] = relu(bn2( Wt @ H[n] + bias ) [+res])
// Wt: (CO, CI); H: (N, CI, W); res: (N, CO, W) or nullptr.
// ---------------------------------------------------------------------------
__global__ __launch_bounds__(32) void tcn_wmma(
    const float* __restrict__ H, const float* __restrict__ Wt,
    const float* __restrict__ bias, const float* __restrict__ bg,
    const float* __restrict__ bb, const float* __restrict__ bm,
    const float* __restrict__ bv, const float* __restrict__ res,
    float* __restrict__ out, int CI, int CO, int W, int NT) {
  int n = blockIdx.x;
  int mt = blockIdx.y / NT, nt = blockIdx.y % NT;
  int lane = threadIdx.x & 31;
  int l16 = lane & 15;
  int hi = lane >> 4;
  int kh = hi * 2;
  int o = mt * 16 + l16;
  int col = nt * 16 + l16;
  bool cok = col < W;
  int colc = cok ? col : 0;
  v8f acc = {};
  for (int kk = 0; kk < CI; kk += 4) {
    v2f a, b;
    a.x = Wt[o * CI + kk + kh];
    a.y = Wt[o * CI + kk + kh + 1];
    float b0 = H[((long)n * CI + kk + kh) * W + colc];
    float b1 = H[((long)n * CI + kk + kh + 1) * W + colc];
    b.x = cok ? b0 : 0.f;
    b.y = cok ? b1 : 0.f;
    acc = __builtin_amdgcn_wmma_f32_16x16x4_f32(false, a, false, b, (short)0, acc,
                                                false, false);
  }
  if (cok) {
    for (int r = 0; r < 8; ++r) {
      int o2 = mt * 16 + r + 8 * hi;
      float z = acc[r] + bias[o2];
      z = bnf(z, bg, bb, bm, bv, o2);
      if (res) z += res[((long)n * CO + o2) * W + col];
      out[((long)n * CO + o2) * W + col] = fmaxf(z, 0.f);
    }
  }
}

// ---------------------------------------------------------------------------
// Transformer (L=5, heads=2), one sequence per thread, scalar fp32.
// in/out layout: (B, 5, D) flat.
// ---------------------------------------------------------------------------
template <int D>
__device__ __forceinline__ void ln_row(const float* src, float* dst, const float* g,
                                       const float* b) {
  float mu = 0.f;
  for (int d = 0; d < D; ++d) mu += src[d];
  mu *= (1.f / D);
  float var = 0.f;
  for (int d = 0; d < D; ++d) {
    float t = src[d] - mu;
    var += t * t;
  }
  var *= (1.f / D);
  float inv = rsqrtf(var + EPS);
  for (int d = 0; d < D; ++d) dst[d] = (src[d] - mu) * inv * g[d] + b[d];
}

template <int D>
__global__ void trans_kernel(const float* __restrict__ in, float* __restrict__ outp,
                             const float* __restrict__ pos, const float* ln1g,
                             const float* ln1b, const float* qkvw, const float* qkvb,
                             const float* ow, const float* ob, const float* ln2g,
                             const float* ln2b, const float* fc1w, const float* fc1b,
                             const float* fc2w, const float* fc2b, const float* lnfg,
                             const float* lnfb, int B) {
  constexpr int L = 5;
  constexpr int DH = D / 2;
  int bi = blockIdx.x * blockDim.x + threadIdx.x;
  if (bi >= B) return;
  const float* xin = in + (long)bi * L * D;
  float x[L * D], h[L * D];
  for (int i = 0; i < L * D; ++i) x[i] = xin[i] + pos[i];
  for (int l = 0; l < L; ++l) ln_row<D>(&x[l * D], &h[l * D], ln1g, ln1b);
  // qkv
  float q[L * D], k_[L * D], v_[L * D];
  for (int l = 0; l < L; ++l) {
#pragma unroll 1
    for (int j = 0; j < 3 * D; ++j) {
      float s = qkvb[j];
      for (int i = 0; i < D; ++i) s += h[l * D + i] * qkvw[i * 3 * D + j];
      if (j < D) q[l * D + j] = s;
      else if (j < 2 * D) k_[l * D + j - D] = s;
      else v_[l * D + j - 2 * D] = s;
    }
  }
  // attention (2 heads)
  float o[L * D];
  float scale = rsqrtf((float)DH);
  for (int hd = 0; hd < 2; ++hd) {
    int off = hd * DH;
    for (int l = 0; l < L; ++l) {
      float s[L];
      float mx = -3.0e38f;
      for (int mm = 0; mm < L; ++mm) {
        float t = 0.f;
        for (int d = 0; d < DH; ++d) t += q[l * D + off + d] * k_[mm * D + off + d];
        t *= scale;
        s[mm] = t;
        mx = fmaxf(mx, t);
      }
      float sum = 0.f;
      for (int mm = 0; mm < L; ++mm) {
        s[mm] = expf(s[mm] - mx);
        sum += s[mm];
      }
      float invs = 1.f / sum;
      for (int d = 0; d < DH; ++d) {
        float a = 0.f;
        for (int mm = 0; mm < L; ++mm) a += s[mm] * v_[mm * D + off + d];
        o[l * D + off + d] = a * invs;
      }
    }
  }
  // out proj + residual
  for (int l = 0; l < L; ++l) {
#pragma unroll 1
    for (int j = 0; j < D; ++j) {
      float s = ob[j];
      for (int i = 0; i < D; ++i) s += o[l * D + i] * ow[i * D + j];
      x[l * D + j] += s;
    }
  }
  // MLP
  for (int l = 0; l < L; ++l) ln_row<D>(&x[l * D], &h[l * D], ln2g, ln2b);
  for (int l = 0; l < L; ++l) {
    float acc[D];
    for (int j = 0; j < D; ++j) acc[j] = fc2b[j];
#pragma unroll 1
    for (int jj = 0; jj < 2 * D; ++jj) {
      float t = fc1b[jj];
      for (int i = 0; i < D; ++i) t += h[l * D + i] * fc1w[i * 2 * D + jj];
      float u = 0.7978845608028654f * (t + 0.044715f * t * t * t);
      float ge = 0.5f * t * (1.f + tanhf(u));
      for (int j = 0; j < D; ++j) acc[j] += ge * fc2w[jj * D + j];
    }
    for (int j = 0; j < D; ++j) x[l * D + j] += acc[j];
  }
  // final LN
  float* op = outp + (long)bi * L * D;
  for (int l = 0; l < L; ++l) ln_row<D>(&x[l * D], &op[l * D], lnfg, lnfb);
}

// ---------------------------------------------------------------------------
// K6: g1 gcn (16->128 with gather from t0 output) + residual branch (16->32)
// ---------------------------------------------------------------------------
__global__ void gcn1_kernel(const float* __restrict__ T0, const float* __restrict__ w,
                            const float* __restrict__ bias, const float* __restrict__ rw,
                            const float* __restrict__ rb2, const float* rg,
                            const float* rbb, const float* rm, const float* rv,
                            float* __restrict__ Y1, float* __restrict__ R1) {
  int i = blockIdx.x * blockDim.x + threadIdx.x;
  if (i >= 1024 * 85) return;
  int n = i / 85, tv = i % 85;
  int t = tv / 17, vv = tv % 17;
  float xin[16];
  const float* src = T0 + ((long)(n * 17 + vv)) * 80 + t * 16;
  for (int ci = 0; ci < 16; ++ci) xin[ci] = src[ci];
  for (int o = 0; o < 128; ++o) {
    float s = bias[o];
    for (int ci = 0; ci < 16; ++ci) s += xin[ci] * w[o * 16 + ci];
    int k = o >> 5, c = o & 31;
    Y1[((long)(n * 32 + c)) * 340 + k * 85 + tv] = s;
  }
  for (int c = 0; c < 32; ++c) {
    float s = rb2[c];
    for (int ci = 0; ci < 16; ++ci) s += xin[ci] * rw[c * 16 + ci];
    R1[((long)(n * 32 + c)) * 85 + tv] = bnf(s, rg, rbb, rm, rv, c);
  }
}

// ---------------------------------------------------------------------------
// K9a/K14a: mean over gsz contiguous elements (raw-reshape mean)
// ---------------------------------------------------------------------------
__global__ void mean_kernel(const float* __restrict__ X, float* __restrict__ out,
                            int total, int groups, int gsz) {
  int i = blockIdx.x * blockDim.x + threadIdx.x;
  if (i >= total) return;
  int n = i / groups, j = i % groups;
  const float* p = X + (long)n * groups * gsz + (long)j * gsz;
  float s = 0.f;
  for (int u = 0; u < gsz; ++u) s += p[u];
  out[i] = s / (float)gsz;
}

// K9b: body-part max pooling: SX[n][c*25+t*5+p] = max_{v in PART[p]} X2[n][c][t*17+v]
__global__ void maxpart_kernel(const float* __restrict__ X2, float* __restrict__ SX) {
  int i = blockIdx.x * blockDim.x + threadIdx.x;
  if (i >= 1024 * 800) return;
  int p = i % 5, t = (i / 5) % 5, c = (i / 25) % 32, n = i / 800;
  float mx = -3.0e38f;
  for (int u = c_part_off[p]; u < c_part_off[p + 1]; ++u)
    mx = fmaxf(mx, X2[(long)n * 2720 + c * 85 + t * 17 + c_part_idx[u]]);
  SX[i] = mx;
}

// ---------------------------------------------------------------------------
// K11: gp gcn (32->256, gather from t1 output) + residual (32->64)
// ---------------------------------------------------------------------------
__global__ void gcnp_kernel(const float* __restrict__ T1, const float* __restrict__ w,
                            const float* __restrict__ bias, const float* __restrict__ rw,
                            const float* __restrict__ rb2, const float* rg,
                            const float* rbb, const float* rm, const float* rv,
                            float* __restrict__ Yp, float* __restrict__ R2) {
  int i = blockIdx.x * blockDim.x + threadIdx.x;
  if (i >= 1024 * 25) return;
  int n = i / 25, tp = i % 25;
  int t = tp / 5, p = tp % 5;
  float xin[32];
  const float* src = T1 + ((long)(n * 5 + p)) * 160 + t * 32;
  for (int ci = 0; ci < 32; ++ci) xin[ci] = src[ci];
  for (int o = 0; o < 256; ++o) {
    float s = bias[o];
    for (int ci = 0; ci < 32; ++ci) s += xin[ci] * w[o * 32 + ci];
    int k = o >> 6, c = o & 63;
    Yp[((long)(n * 64 + c)) * 100 + k * 25 + tp] = s;
  }
  for (int c = 0; c < 64; ++c) {
    float s = rb2[c];
    for (int ci = 0; ci < 32; ++ci) s += xin[ci] * rw[c * 32 + ci];
    R2[((long)(n * 64 + c)) * 25 + tp] = bnf(s, rg, rbb, rm, rv, c);
  }
}

// K14b: xp1 = max over p: XS2[n][c*5+t] = max_p XP[n][c*25+t*5+p]
__global__ void maxnp_kernel(const float* __restrict__ XP, float* __restrict__ XS2) {
  int i = blockIdx.x * blockDim.x + threadIdx.x;
  if (i >= 1024 * 320) return;
  int t = i % 5, c = (i / 5) % 64, n = i / 320;
  float mx = -3.0e38f;
  for (int p = 0; p < 5; ++p)
    mx = fmaxf(mx, XP[(long)n * 1600 + c * 25 + t * 5 + p]);
  XS2[i] = mx;
}

// K16: c4 temporal conv (k=3, pad 1) on t2 output + bn + relu
__global__ void c4_kernel(const float* __restrict__ T2, const float* __restrict__ w,
                          const float* __restrict__ bias, const float* g, const float* b,
                          const float* m, const float* v, float* __restrict__ XC4) {
  int i = blockIdx.x * blockDim.x + threadIdx.x;
  if (i >= 1024 * 640) return;
  int t = i % 5, o = (i / 5) % 128, n = i / 640;
  float s = bias[o];
  for (int ci = 0; ci < 64; ++ci) {
    for (int r = 0; r < 3; ++r) {
      int tt = t + r - 1;
      if (tt < 0 || tt >= 5) continue;
      s += T2[(long)n * 320 + tt * 64 + ci] * w[o * 192 + ci * 3 + r];
    }
  }
  s = bnf(s, g, b, m, v, o);
  XC4[i] = fmaxf(s, 0.f);
}

// K16b: Rm[n][ch][i] = sum_m xap2[n][i*5+m] * XC4[n][ch][m]
__global__ void rm_kernel(const float* __restrict__ xap2, const float* __restrict__ XC4,
                          float* __restrict__ Rm) {
  int i = blockIdx.x * blockDim.x + threadIdx.x;
  if (i >= 1024 * 640) return;
  int t = i % 5, ch = (i / 5) % 128, n = i / 640;
  float s = 0.f;
  for (int mm = 0; mm < 5; ++mm)
    s += xap2[n * 25 + t * 5 + mm] * XC4[(long)n * 640 + ch * 5 + mm];
  Rm[i] = s;
}

// K17: c2 1x1 conv 128->256 + bn + relu (j-independent)
__global__ void c2_kernel(const float* __restrict__ Rm, const float* __restrict__ w,
                          const float* __restrict__ bias, const float* g, const float* b,
                          const float* m, const float* v, float* __restrict__ U) {
  int i = blockIdx.x * blockDim.x + threadIdx.x;
  if (i >= 1024 * 1280) return;
  int t = i % 5, oc = (i / 5) % 256, n = i / 1280;
  float s = bias[oc];
  for (int ch = 0; ch < 128; ++ch) s += w[oc * 128 + ch] * Rm[(long)n * 640 + ch * 5 + t];
  U[i] = fmaxf(bnf(s, g, b, m, v, oc), 0.f);
}

// K18: final fusion: out[n,o3,i,v] = fcn_b[o3] + S1[n][i]*G[n][o3][J2P[v]]
__global__ void final_kernel(const float* __restrict__ U, const float* __restrict__ xap1,
                             const float* __restrict__ fw, const float* __restrict__ fb,
                             float* __restrict__ out) {
  int n = blockIdx.x;
  int tid = threadIdx.x;
  __shared__ float G[15];
  __shared__ float S1[5];
  if (tid < 15) {
    int o3 = tid / 5, p = tid % 5;
    float s = 0.f;
    for (int oc = 0; oc < 256; ++oc) s += fw[o3 * 256 + oc] * U[(long)n * 1280 + oc * 5 + p];
    G[tid] = s;
  }
  if (tid >= 32 && tid < 37) {
    int ii = tid - 32;
    float s = 0.f;
    for (int vv = 0; vv < 17; ++vv) s += xap1[n * 85 + ii * 17 + vv];
    S1[ii] = s;
  }
  __syncthreads();
  if (tid < 255) {
    int o3 = tid / 85, r = tid % 85, ii = r / 17, vv = r % 17;
    out[(long)n * 255 + tid] = fb[o3] + S1[ii] * G[o3 * 5 + c_j2p[vv]];
  }
}

// ============================================================================
// Workspace layout (float offsets), lifetimes hand-checked for reuse.
// ============================================================================
static const long OFF_X0 = 0;          // 174080
static const long OFF_Y0 = 174080;     // 5570560
static const long OFF_H0 = 5744640;    // 1392640
static const long OFF_X1 = 0;          // 1392640 (Y0 dead)
static const long OFF_T0 = 7137280;    // 1392640
static const long OFF_Y1 = 8529920;    // 11141120
static const long OFF_R1 = 0;          // 2785280 (X1 dead)
static const long OFF_H1 = 2785280;    // 2785280
static const long OFF_X2 = 5570560;    // 2785280
static const long OFF_AP1 = 19671040;  // 87040  (persistent)
static const long OFF_AP2 = 19758080;  // 25600  (persistent)
static const long OFF_SX = 0;          // 819200
static const long OFF_T1 = 819200;     // 819200
static const long OFF_YP = 1638400;    // 6553600
static const long OFF_R2 = 8192000;    // 1638400
static const long OFF_HP = 9830400;    // 1638400
static const long OFF_XP = 11468800;   // 1638400
static const long OFF_XS2 = 0;         // 327680
static const long OFF_T2 = 327680;     // 327680
static const long OFF_XC4 = 655360;    // 655360
static const long OFF_RM = 1310720;    // 655360
static const long OFF_U = 1966080;     // 1310720
// total requirement: 19,783,680 floats = 79.2 MB

extern "C" void kernel_launch(void* const* d_in, const int* in_sizes, int n_in,
                              void* d_out, int out_size, void* d_ws, size_t ws_size,
                              hipStream_t stream) {
  (void)in_sizes; (void)n_in; (void)out_size; (void)ws_size;
  auto P = [&](int i) { return (const float*)d_in[i]; };
  float* ws = (float*)d_ws;
  float* out = (float*)d_out;
  // input indices (setup_inputs dict order, nested dicts flattened in order):
  // 0 x, 1 A, 2 A_pool, 3-6 bn0{g,b,m,v},
  // g0: 7 gcn_w 8 gcn_b 9-12 bn1 13 tcn_w 14 tcn_b 15-18 bn2
  // g1: 19 gcn_w 20 gcn_b 21-24 bn1 25 tcn_w 26 tcn_b 27-30 bn2 31 res_w 32 res_b 33-36 res_bn
  // gp: 37 gcn_w 38 gcn_b 39-42 bn1 43 tcn_w 44 tcn_b 45-48 bn2 49 res_w 50 res_b 51-54 res_bn
  // t0: 55..69  t1: 70..84  t2: 85..99
  // 100 c4_w 101 c4_b 102-105 c4_bn, 106 c2_w 107 c2_b 108-111 c2_bn, 112 fcn_w 113 fcn_b

  bn0_kernel<<<680, 256, 0, stream>>>(P(0), P(3), P(4), P(5), P(6), ws + OFF_X0);
  gcn0_kernel<<<340, 256, 0, stream>>>(ws + OFF_X0, P(7), P(8), ws + OFF_Y0);
  einsum_wmma<<<dim3(1024, 6), 32, 0, stream>>>(ws + OFF_Y0, P(1), P(9), P(10), P(11),
                                                P(12), ws + OFF_H0, 16, 340, 85, 6);
  tcn_wmma<<<dim3(1024, 6), 32, 0, stream>>>(ws + OFF_H0, P(13), P(14), P(15), P(16),
                                             P(17), P(18), nullptr, ws + OFF_X1, 16, 16,
                                             85, 6);
  trans_kernel<16><<<136, 128, 0, stream>>>(ws + OFF_X1, ws + OFF_T0, P(55), P(56),
                                            P(57), P(58), P(59), P(60), P(61), P(62),
                                            P(63), P(64), P(65), P(66), P(67), P(68),
                                            P(69), 17408);
  gcn1_kernel<<<340, 256, 0, stream>>>(ws + OFF_T0, P(19), P(20), P(31), P(32), P(33),
                                       P(34), P(35), P(36), ws + OFF_Y1, ws + OFF_R1);
  einsum_wmma<<<dim3(1024, 12), 32, 0, stream>>>(ws + OFF_Y1, P(1), P(21), P(22), P(23),
                                                 P(24), ws + OFF_H1, 32, 340, 85, 6);
  tcn_wmma<<<dim3(1024, 12), 32, 0, stream>>>(ws + OFF_H1, P(25), P(26), P(27), P(28),
                                              P(29), P(30), ws + OFF_R1, ws + OFF_X2, 32,
                                              32, 85, 6);
  mean_kernel<<<340, 256, 0, stream>>>(ws + OFF_X2, ws + OFF_AP1, 1024 * 85, 85, 32);
  maxpart_kernel<<<3200, 256, 0, stream>>>(ws + OFF_X2, ws + OFF_SX);
  trans_kernel<32><<<40, 128, 0, stream>>>(ws + OFF_SX, ws + OFF_T1, P(70), P(71), P(72),
                                           P(73), P(74), P(75), P(76), P(77), P(78),
                                           P(79), P(80), P(81), P(82), P(83), P(84),
                                           5120);
  gcnp_kernel<<<100, 256, 0, stream>>>(ws + OFF_T1, P(37), P(38), P(49), P(50), P(51),
                                       P(52), P(53), P(54), ws + OFF_YP, ws + OFF_R2);
  einsum_wmma<<<dim3(1024, 8), 32, 0, stream>>>(ws + OFF_YP, P(2), P(39), P(40), P(41),
                                                P(42), ws + OFF_HP, 64, 100, 25, 2);
  tcn_wmma<<<dim3(1024, 8), 32, 0, stream>>>(ws + OFF_HP, P(43), P(44), P(45), P(46),
                                             P(47), P(48), ws + OFF_R2, ws + OFF_XP, 64,
                                             64, 25, 2);
  mean_kernel<<<100, 256, 0, stream>>>(ws + OFF_XP, ws + OFF_AP2, 1024 * 25, 25, 64);
  maxnp_kernel<<<1280, 256, 0, stream>>>(ws + OFF_XP, ws + OFF_XS2);
  trans_kernel<64><<<8, 128, 0, stream>>>(ws + OFF_XS2, ws + OFF_T2, P(85), P(86), P(87),
                                          P(88), P(89), P(90), P(91), P(92), P(93),
                                          P(94), P(95), P(96), P(97), P(98), P(99),
                                          1024);
  c4_kernel<<<2560, 256, 0, stream>>>(ws + OFF_T2, P(100), P(101), P(102), P(103),
                                      P(104), P(105), ws + OFF_XC4);
  rm_kernel<<<2560, 256, 0, stream>>>(ws + OFF_AP2, ws + OFF_XC4, ws + OFF_RM);
  c2_kernel<<<5120, 256, 0, stream>>>(ws + OFF_RM, P(106), P(107), P(108), P(109),
                                      P(110), P(111), ws + OFF_U);
  final_kernel<<<1024, 256, 0, stream>>>(ws + OFF_U, ws + OFF_AP1, P(112), P(113), out);
}